// Fourier3dFunctionCUDA_57629871178216
// MI455X (gfx1250) — compile-verified
//
#include <hip/hip_runtime.h>

typedef __attribute__((ext_vector_type(2))) float v2f;
typedef __attribute__((ext_vector_type(8))) float v8f;

// Fast hardware reciprocal (v_rcp_f32, ~1 ulp) instead of IEEE-div expansion.
__device__ __forceinline__ float fast_rcp(float x) {
    return __builtin_amdgcn_rcpf(x);
}

// ---------------------------------------------------------------------------
// Prep: for each face f, stage 4 row-vectors {v0, v1, v2, N=cross(e1,e2)} as
// [face][4 rows][4 floats (x,y,z,0)] -> WMMA A-matrix staging:
// group g = 4 faces -> 16 rows x K=4.
// ---------------------------------------------------------------------------
__global__ void mesh_ft_prep(const float* __restrict__ Verts,
                             const int* __restrict__ Faces,
                             float* __restrict__ rows,
                             int nFaces, int nFacesPadded)
{
    int f = blockIdx.x * blockDim.x + threadIdx.x;
    if (f >= nFacesPadded) return;
    float* r = rows + (size_t)f * 16;
    if (f >= nFaces) {
#pragma unroll
        for (int i = 0; i < 16; ++i) r[i] = 0.0f;
        return;
    }
    int ia = Faces[3 * f + 0], ib = Faces[3 * f + 1], ic = Faces[3 * f + 2];
    float ax = Verts[3 * ia + 0], ay = Verts[3 * ia + 1], az = Verts[3 * ia + 2];
    float bx = Verts[3 * ib + 0], by = Verts[3 * ib + 1], bz = Verts[3 * ib + 2];
    float cx = Verts[3 * ic + 0], cy = Verts[3 * ic + 1], cz = Verts[3 * ic + 2];
    float e1x = bx - ax, e1y = by - ay, e1z = bz - az;
    float e2x = cx - ax, e2y = cy - ay, e2z = cz - az;
    float nx = e1y * e2z - e1z * e2y;
    float ny = e1z * e2x - e1x * e2z;
    float nz = e1x * e2y - e1y * e2x;
    r[0]  = ax; r[1]  = ay; r[2]  = az; r[3]  = 0.0f;
    r[4]  = bx; r[5]  = by; r[6]  = bz; r[7]  = 0.0f;
    r[8]  = cx; r[9]  = cy; r[10] = cz; r[11] = 0.0f;
    r[12] = nx; r[13] = ny; r[14] = nz; r[15] = 0.0f;
}

// ---------------------------------------------------------------------------
// Main: one wave32 per 16-frequency tile. Per face group (4 faces) one
// v_wmma_f32_16x16x4_f32 computes u_j = xi . v_j and xi . N for
// 16 frequencies x 4 faces; each lane then holds complete quadruples
// (u0,u1,u2,xiN) for 2 faces of its own frequency column and does the
// complex divided-difference locally. Lane halves are folded with shfl_xor.
// ---------------------------------------------------------------------------
__global__ __launch_bounds__(256)
void mesh_ft_main(const float* __restrict__ rows,
                  const float* __restrict__ xi0,
                  const float* __restrict__ xi1,
                  const float* __restrict__ xi2,
                  float* __restrict__ out,
                  int nFaces, int nGroups,
                  int n1, int n2, int M)
{
    extern __shared__ float smem[];
    const int tid = threadIdx.x;

    // Cooperative stage of all face rows into LDS (nGroups * 16 float4s).
    {
        const float4* src = (const float4*)rows;
        float4* dst = (float4*)smem;
        const int n4 = nGroups * 16;
        for (int i = tid; i < n4; i += blockDim.x) dst[i] = src[i];
    }
    __syncthreads();

    const int lane    = tid & 31;
    const int wave    = tid >> 5;
    const int tile    = blockIdx.x * (blockDim.x >> 5) + wave;
    const int col     = lane & 15;         // frequency column within tile
    const bool hiHalf = lane >= 16;
    const int gfreq   = tile * 16 + col;

    // Frequency for this lane's column (ij-meshgrid ravel order).
    const int gq = (gfreq < M) ? gfreq : (M - 1);
    const int i2 = gq % n2;
    const int i1 = (gq / n2) % n1;
    const int i0 = gq / (n2 * n1);
    const float fx = xi0[i0], fy = xi1[i1], fz = xi2[i2];
    const float xi_sq = fx * fx + fy * fy + fz * fz;

    // B fragment (4x16, K split across lane halves): lo = (x,y), hi = (z,0).
    v2f bfrag;
    bfrag.x = hiHalf ? fz : fx;
    bfrag.y = hiHalf ? 0.0f : fy;

    float Sr = 0.0f, Si = 0.0f;
    const int rowSel = col * 4 + (hiHalf ? 2 : 0);  // A fragment offset in a group

    // Software pipeline: load group g+1's fragment before consuming group g.
    v2f afrag = *(const v2f*)(smem + rowSel);
    for (int g = 0; g < nGroups; ++g) {
        v2f anext = afrag;
        if (g + 1 < nGroups)
            anext = *(const v2f*)(smem + (g + 1) * 64 + rowSel);

        v8f cacc = {};
        // D = A(16x4 face rows) x B(4x16 freqs)
        v8f d = __builtin_amdgcn_wmma_f32_16x16x4_f32(
            false, afrag, false, bfrag, (short)0, cacc, false, false);
        afrag = anext;

        // lo-half lanes: D rows 0..7  -> faces 4g+0, 4g+1
        // hi-half lanes: D rows 8..15 -> faces 4g+2, 4g+3
        const int faceBase = 4 * g + (hiHalf ? 2 : 0);
#pragma unroll
        for (int j = 0; j < 2; ++j) {
            float u0 = d[4 * j + 0];
            float u1 = d[4 * j + 1];
            float u2 = d[4 * j + 2];
            float xN = d[4 * j + 3];
            float d01 = u0 - u1, d02 = u0 - u2, d12 = u1 - u2;
            // One rcp instead of three: invP = 1/(d01*d02*d12), then
            //   1/(d01*d02) = invP*d12,  1/(d01*d12) = invP*d02,  1/(d02*d12) = invP*d01
            float invP   = fast_rcp(d01 * d02 * d12);
            float r01_02 = invP * d12;
            float r01_12 = invP * d02;
            float r02_12 = invP * d01;
            float s0, c0, s1, c1, s2, c2;
            __sincosf(u0, &s0, &c0);
            __sincosf(u1, &s1, &c1);
            __sincosf(u2, &s2, &c2);
            // E_j = exp(-i u_j) = cos u_j - i sin u_j
            // dd = E0/(d01*d02) - E1/(d01*d12) + E2/(d02*d12); S = sum xN*(-dd)
            float ddr =  c0 * r01_02 - c1 * r01_12 + c2 * r02_12;
            float ddi = -s0 * r01_02 + s1 * r01_12 - s2 * r02_12;
            bool valid = (faceBase + j) < nFaces;   // branch-free: keep EXEC all-1s
            Sr -= valid ? xN * ddr : 0.0f;
            Si -= valid ? xN * ddi : 0.0f;
        }
    }

    // Fold the two lane halves (each holds half the faces of the same freq).
    Sr += __shfl_xor(Sr, 16, 32);
    Si += __shfl_xor(Si, 16, 32);

    if (!hiHalf && gfreq < M) {
        float inv = fast_rcp(xi_sq);
        // ft = i * S / |xi|^2  ->  re = -Si/|xi|^2, im = Sr/|xi|^2
        out[2 * (size_t)gfreq + 0] = -Si * inv;
        out[2 * (size_t)gfreq + 1] =  Sr * inv;
    }
}

extern "C" void kernel_launch(void* const* d_in, const int* in_sizes, int n_in,
                              void* d_out, int out_size, void* d_ws, size_t ws_size,
                              hipStream_t stream)
{
    const float* Verts = (const float*)d_in[0];
    const int*   Faces = (const int*)d_in[1];
    const float* xi0   = (const float*)d_in[2];
    const float* xi1   = (const float*)d_in[3];
    const float* xi2   = (const float*)d_in[4];
    float* out = (float*)d_out;   // complex64 interleaved (re, im)

    const int nFaces = in_sizes[1] / 3;
    const int n0 = in_sizes[2], n1 = in_sizes[3], n2 = in_sizes[4];
    const int M = n0 * n1 * n2;
    const int nGroups = (nFaces + 3) / 4;
    const int nFacesPadded = nGroups * 4;

    float* rows = (float*)d_ws;   // staging: nFacesPadded * 16 floats

    {
        const int threads = 256;
        const int blocks = (nFacesPadded + threads - 1) / threads;
        hipLaunchKernelGGL(mesh_ft_prep, dim3(blocks), dim3(threads), 0, stream,
                           Verts, Faces, rows, nFaces, nFacesPadded);
    }
    {
        const int threads = 256;                  // 8 waves (wave32)
        const int wavesPerBlock = threads / 32;
        const int tiles = (M + 15) / 16;
        const int blocks = (tiles + wavesPerBlock - 1) / wavesPerBlock;
        const size_t smemBytes = (size_t)nGroups * 64 * sizeof(float);
        hipLaunchKernelGGL(mesh_ft_main, dim3(blocks), dim3(threads), smemBytes, stream,
                           rows, xi0, xi1, xi2, out, nFaces, nGroups, n1, n2, M);
    }
}